// Net_43679817401125
// MI455X (gfx1250) — compile-verified
//
#include <hip/hip_runtime.h>
#include <hip/hip_bf16.h>
#include <math.h>

typedef __attribute__((ext_vector_type(2))) float v2f;
typedef __attribute__((ext_vector_type(8))) float v8f;

#define HIDDEN 64
#define NCLS   16
#define NREL   16

// ---------------- zero workspace ----------------
__global__ void zero_f32(float* __restrict__ p, long n) {
  long i = (long)blockIdx.x * blockDim.x + threadIdx.x;
  if (i < n) p[i] = 0.0f;
}

// ---------------- layer-1 edge gather + scatter-add ----------------
// one wave (32 lanes) per edge; each lane handles channels {lane, lane+32}
__global__ void rgcn_scatter1(const int* __restrict__ edge_index,
                              const int* __restrict__ edge_type,
                              const float* __restrict__ W1,
                              float* __restrict__ agg1,
                              int nEdges, int nNodes) {
  long tid = (long)blockIdx.x * blockDim.x + threadIdx.x;
  long e = tid >> 5;
  int lane = (int)(tid & 31);
  if (e >= nEdges) return;
  int s = edge_index[e];
  int d = edge_index[(long)nEdges + e];
  int r = edge_type[e];
  const float* wrow = W1 + ((size_t)r * (size_t)nNodes + (size_t)s) * HIDDEN;
  float* arow = agg1 + (size_t)d * HIDDEN;
  float v0 = wrow[lane];
  float v1 = wrow[lane + 32];
  atomicAdd(arow + lane, v0);
  atomicAdd(arow + lane + 32, v1);
}

// ---------------- x = relu(agg1 + root1 + b1) ----------------
__global__ void rgcn_relu(float* __restrict__ x, const float* __restrict__ root1,
                          const float* __restrict__ b1, long n) {
  long i = (long)blockIdx.x * blockDim.x + threadIdx.x;
  if (i < n) {
    float v = x[i] + root1[i] + b1[i & (HIDDEN - 1)];
    x[i] = v > 0.0f ? v : 0.0f;
  }
}

// ---------------- grouped GEMM via native fp32 WMMA ----------------
// wave w computes one 16x16 tile of h[rel] = x @ W2[rel]  (rel = 0..15)
// or, for rel==16, one tile of out = x @ root2 + b2 (initializes d_out).
// K=64 swept in 16 steps of V_WMMA_F32_16X16X4_F32.
__global__ void rgcn_gemm_wmma(const float* __restrict__ x,      // [N,64]
                               const float* __restrict__ W2,     // [16,64,16]
                               const float* __restrict__ root2,  // [64,16]
                               const float* __restrict__ b2,     // [16]
                               float* __restrict__ h,            // [16,N,16]
                               float* __restrict__ out,          // [N,16]
                               int nNodes) {
  int wid  = (int)(((long)blockIdx.x * blockDim.x + threadIdx.x) >> 5);
  int lane = threadIdx.x & 31;
  int mtiles = (nNodes + 15) >> 4;
  int rel = wid / mtiles;
  int mt  = wid - rel * mtiles;
  if (rel > NREL) return;   // wave-uniform: EXEC stays all-1s in active waves

  const float* B = (rel < NREL) ? (W2 + (size_t)rel * HIDDEN * NCLS) : root2;

  // A-fragment addressing: row = mt*16 + (lane&15); lane-half selects K pair
  int arow = mt * 16 + (lane & 15);
  if (arow >= nNodes) arow = nNodes - 1;       // clamp; invalid rows masked on store
  int kh = (lane >> 4) << 1;                   // 0 (lanes 0-15) or 2 (lanes 16-31)
  int n  = lane & 15;
  const float* xr = x + (size_t)arow * HIDDEN;

  v8f c = {};
#pragma unroll
  for (int kk = 0; kk < HIDDEN; kk += 4) {
    v2f a, b;
    a.x = xr[kk + kh];
    a.y = xr[kk + kh + 1];
    b.x = B[(kk + kh) * NCLS + n];
    b.y = B[(kk + kh + 1) * NCLS + n];
    // D = A(16x4 f32) * B(4x16 f32) + C(16x16 f32)
    c = __builtin_amdgcn_wmma_f32_16x16x4_f32(false, a, false, b,
                                              (short)0, c, false, false);
  }

  // D layout: VGPR v, lane l -> M = v + 8*(l>=16), N = l&15
  int mbase = mt * 16 + ((lane >> 4) << 3);
  if (rel < NREL) {
    float* hp = h + (size_t)rel * (size_t)nNodes * NCLS;
#pragma unroll
    for (int v = 0; v < 8; ++v) {
      int m = mbase + v;
      if (m < nNodes) hp[(size_t)m * NCLS + n] = c[v];
    }
  } else {
    float bias = b2[n];
#pragma unroll
    for (int v = 0; v < 8; ++v) {
      int m = mbase + v;
      if (m < nNodes) out[(size_t)m * NCLS + n] = c[v] + bias;
    }
  }
}

// ---------------- layer-2 edge gather + scatter-add into out ----------------
// 16 lanes per edge (one per class)
__global__ void rgcn_scatter2(const int* __restrict__ edge_index,
                              const int* __restrict__ edge_type,
                              const float* __restrict__ h,
                              float* __restrict__ out,
                              int nEdges, int nNodes) {
  long tid = (long)blockIdx.x * blockDim.x + threadIdx.x;
  long e = tid >> 4;
  int c = (int)(tid & 15);
  if (e >= nEdges) return;
  int s = edge_index[e];
  int d = edge_index[(long)nEdges + e];
  int r = edge_type[e];
  float v = h[((size_t)r * (size_t)nNodes + (size_t)s) * NCLS + c];
  atomicAdd(out + (size_t)d * NCLS + c, v);
}

// ---------------- in-place row log-softmax ----------------
__global__ void rgcn_logsoftmax(float* __restrict__ out, int nNodes) {
  int node = blockIdx.x * blockDim.x + threadIdx.x;
  if (node >= nNodes) return;
  float* row = out + (size_t)node * NCLS;
  float v[NCLS];
  float mx = -3.4e38f;
#pragma unroll
  for (int c = 0; c < NCLS; ++c) { v[c] = row[c]; mx = fmaxf(mx, v[c]); }
  float s = 0.0f;
#pragma unroll
  for (int c = 0; c < NCLS; ++c) s += expf(v[c] - mx);
  float ls = logf(s);
#pragma unroll
  for (int c = 0; c < NCLS; ++c) row[c] = v[c] - mx - ls;
}

extern "C" void kernel_launch(void* const* d_in, const int* in_sizes, int n_in,
                              void* d_out, int out_size, void* d_ws, size_t ws_size,
                              hipStream_t stream) {
  const int*   edge_index = (const int*)d_in[0];   // [2,E]
  const int*   edge_type  = (const int*)d_in[1];   // [E] (sorted)
  const float* W1    = (const float*)d_in[2];      // [R,N,64]
  const float* root1 = (const float*)d_in[3];      // [N,64]
  const float* b1    = (const float*)d_in[4];      // [64]
  const float* W2    = (const float*)d_in[5];      // [R,64,16]
  const float* root2 = (const float*)d_in[6];      // [64,16]
  const float* b2    = (const float*)d_in[7];      // [16]
  float* out = (float*)d_out;                      // [N,16]

  const int nEdges = in_sizes[1];
  const int nNodes = in_sizes[3] / HIDDEN;

  // workspace: x (=agg1) [N,64] then h [16,N,16]  -> ~32 MB total
  float* x = (float*)d_ws;
  float* h = x + (size_t)nNodes * HIDDEN;

  long nx = (long)nNodes * HIDDEN;
  zero_f32<<<(int)((nx + 255) / 256), 256, 0, stream>>>(x, nx);

  long t1 = (long)nEdges * 32;
  rgcn_scatter1<<<(int)((t1 + 255) / 256), 256, 0, stream>>>(
      edge_index, edge_type, W1, x, nEdges, nNodes);

  rgcn_relu<<<(int)((nx + 255) / 256), 256, 0, stream>>>(x, root1, b1, nx);

  int mtiles = (nNodes + 15) / 16;
  long waves = (long)(NREL + 1) * mtiles;          // 16 relations + root2 pass
  long t2 = waves * 32;
  rgcn_gemm_wmma<<<(int)((t2 + 255) / 256), 256, 0, stream>>>(
      x, W2, root2, b2, h, out, nNodes);

  long t3 = (long)nEdges * 16;
  rgcn_scatter2<<<(int)((t3 + 255) / 256), 256, 0, stream>>>(
      edge_index, edge_type, h, out, nEdges, nNodes);

  rgcn_logsoftmax<<<(nNodes + 255) / 256, 256, 0, stream>>>(out, nNodes);
}